// SPAIRGlimpseVAE_64269890617428
// MI455X (gfx1250) — compile-verified
//
#include <hip/hip_runtime.h>
#include <math.h>

typedef __attribute__((ext_vector_type(2))) float v2f;
typedef __attribute__((ext_vector_type(8))) float v8f;

// ---------- helpers: monotone uint encoding of float for atomicMax ----------
__device__ __forceinline__ unsigned int f2key(float f) {
  unsigned int b = __float_as_uint(f);
  return (b & 0x80000000u) ? ~b : (b | 0x80000000u);
}
__device__ __forceinline__ float key2f(unsigned int k) {
  unsigned int b = (k & 0x80000000u) ? (k & 0x7FFFFFFFu) : ~k;
  return __uint_as_float(b);
}

// ---------- ws init (key=0 decodes to -NaN => minimum; S=P=0) ----------
__global__ void init_ws_kernel(unsigned int* __restrict__ ws, int n) {
  int i = blockIdx.x * blockDim.x + threadIdx.x;
  if (i < n) ws[i] = 0u;
}

// ---------- pass 1: per-segment max of member_log_mask ----------
__global__ void seg_max_kernel(const float* __restrict__ lm,
                               const int* __restrict__ idx,
                               unsigned int* __restrict__ keys, int M) {
  int i = blockIdx.x * blockDim.x + threadIdx.x;
  if (i < M) {
    atomicMax(&keys[idx[i]], f2key(lm[i]));
  }
}

// ---------- pass 2: S[g] += exp(lm-max), P[g] += pos*exp(lm-max) ----------
__global__ void seg_acc_kernel(const float* __restrict__ lm,
                               const float* __restrict__ pos,
                               const int* __restrict__ idx,
                               const unsigned int* __restrict__ keys,
                               float* __restrict__ S, float* __restrict__ P, int M) {
  int i = blockIdx.x * blockDim.x + threadIdx.x;
  if (i < M) {
    int g = idx[i];
    float w = expf(lm[i] - key2f(keys[g]));
    atomicAdd(&S[g], w);
    atomicAdd(&P[3 * g + 0], pos[3 * (size_t)i + 0] * w);
    atomicAdd(&P[3 * g + 1], pos[3 * (size_t)i + 1] * w);
    atomicAdd(&P[3 * g + 2], pos[3 * (size_t)i + 2] * w);
  }
}

// ---------- head: WMMA f32 GEMV (16 glimpses per wave), tanh/log-sigmoid, fuse centers ----------
// A (16x4 f32) per ISA layout: lanes 0-15 hold {K=0,K=1}, lanes 16-31 hold {K=2,K=3}.
// B (4x16 f32): lanes 0-15 hold rows K=0(V0),K=1(V1); lanes 16-31 rows K=2,K=3.
// We set B[k][n] = W[k] for all n, so every column of D equals the dot product.
// C/D 16x16 f32: VGPR r, lanes 0-15 -> M=r, N=lane; lanes 16-31 -> M=8+r, N=lane-16.
__global__ void head_kernel(const float* __restrict__ F,
                            const float* __restrict__ W,
                            const float* __restrict__ bptr,
                            const float* __restrict__ u,
                            const float* __restrict__ temp,
                            const float* __restrict__ S,
                            const float* __restrict__ P,
                            float* __restrict__ out, int G) {
  const int lane = threadIdx.x & 31;
  const int wave = threadIdx.x >> 5;
  const int tile = blockIdx.x * (blockDim.x >> 5) + wave;
  const int base = tile * 16;
  if (base >= G) return;            // tile uniform across wave: full-wave exit

  const int m  = lane & 15;
  const int hi = lane >> 4;
  int rowg = base + m; if (rowg > G - 1) rowg = G - 1;   // clamp (G%16==0 normally)
  const float* __restrict__ row = F + (size_t)rowg * 256;

  v8f acc = {};
#if __has_builtin(__builtin_amdgcn_wmma_f32_16x16x4_f32)
  for (int k0 = 0; k0 < 256; k0 += 4) {
    v2f a, bb;
    const int ko = k0 + 2 * hi;
    a.x  = row[ko];  a.y  = row[ko + 1];
    bb.x = W[ko];    bb.y = W[ko + 1];
    acc = __builtin_amdgcn_wmma_f32_16x16x4_f32(
        /*neg_a=*/false, a, /*neg_b=*/false, bb,
        /*c_mod=*/(short)0, acc, /*reuse_a=*/false, /*reuse_b=*/false);
  }
#else
  // scalar fallback (keeps round compiling if builtin signature differs)
  float d = 0.f;
  for (int k = 0; k < 256; ++k) d += row[k] * W[k];
  acc[0] = d; acc[1] = d; acc[2] = d; acc[3] = d;
  acc[4] = d; acc[5] = d; acc[6] = d; acc[7] = d;
#endif

  if ((lane & 15) == 0) {           // N==0 holders: lane 0 (rows 0-7), lane 16 (rows 8-15)
    const float bscal = bptr[0];
    const float tinv  = 1.0f / temp[0];
    for (int r = 0; r < 8; ++r) {
      int g = base + 8 * hi + r;
      if (g >= G) break;
      float raw   = acc[r] + bscal;
      float logit = 8.8f * tanhf(raw);
      float uu    = u[g];
      float relaxed = (logit + logf(uu) - log1pf(-uu)) * tinv;
      float logz = (relaxed >= 0.f) ? -log1pf(expf(-relaxed))
                                    : (relaxed - log1pf(expf(relaxed)));
      float s = S[g];
      float inv_s = 1.0f / s;
      float* o = out + (size_t)g * 5;
      o[0] = logz;
      o[1] = logit;
      o[2] = P[3 * g + 0] * inv_s;
      o[3] = P[3 * g + 1] * inv_s;
      o[4] = P[3 * g + 2] * inv_s;
    }
  }
}

extern "C" void kernel_launch(void* const* d_in, const int* in_sizes, int n_in,
                              void* d_out, int out_size, void* d_ws, size_t ws_size,
                              hipStream_t stream) {
  const float* F    = (const float*)d_in[0];   // glimpse_feature (G,256)
  const float* pos  = (const float*)d_in[1];   // member_local_pos (M,3)
  const float* lm   = (const float*)d_in[2];   // member_log_mask (M,1)
  const int*   idx  = (const int*)  d_in[3];   // member_glimpse_index (M,)
  const float* u    = (const float*)d_in[4];   // u (G,)
  const float* W    = (const float*)d_in[5];   // W (256,1)
  const float* b    = (const float*)d_in[6];   // b (1,)
  const float* temp = (const float*)d_in[7];   // temperature scalar
  float* out = (float*)d_out;

  const int G = in_sizes[4];
  const int M = in_sizes[3];

  unsigned int* keys = (unsigned int*)d_ws;          // [0, G)
  float* S = (float*)d_ws + G;                       // [G, 2G)
  float* P = (float*)d_ws + 2 * (size_t)G;           // [2G, 5G)

  const int BS = 256;

  // init workspace (5*G u32 zeroed)
  {
    int n = 5 * G;
    init_ws_kernel<<<(n + BS - 1) / BS, BS, 0, stream>>>((unsigned int*)d_ws, n);
  }
  // pass 1: segment max
  seg_max_kernel<<<(M + BS - 1) / BS, BS, 0, stream>>>(lm, idx, keys, M);
  // pass 2: segment exp-sum + weighted position sum
  seg_acc_kernel<<<(M + BS - 1) / BS, BS, 0, stream>>>(lm, pos, idx, keys, S, P, M);
  // head: WMMA GEMV + activations + fused center normalize
  {
    int tiles = (G + 15) / 16;               // one wave per 16 glimpses
    int wavesPerBlock = BS / 32;             // 8 waves per 256-thread block
    int blocks = (tiles + wavesPerBlock - 1) / wavesPerBlock;
    head_kernel<<<blocks, BS, 0, stream>>>(F, W, b, u, temp, S, P, out, G);
  }
}